// protein_HGNN_18975165514621
// MI455X (gfx1250) — compile-verified
//
#include <hip/hip_runtime.h>
#include <hip/hip_bf16.h>

typedef float v2f __attribute__((ext_vector_type(2)));
typedef float v8f __attribute__((ext_vector_type(8)));

#define C 128            // all channel dims are 128
#define BN_EPS 1e-5f

// ---------------------------------------------------------------------------
// GEMM: Y[M,128] = A[M,128] @ W^T   (W is [128,128] row-major: W[out][in])
// fp32 WMMA 16x16x4, K unrolled 32x. One wave per 16x16 output tile,
// 8 waves (256 thr) per block covering one 16-row stripe across all 128 cols.
// EXEC is all-ones inside the WMMA region (uniform early-out only).
// ---------------------------------------------------------------------------
__global__ __launch_bounds__(256) void k_gemm_f32_wmma(
    const float* __restrict__ A, const float* __restrict__ W,
    float* __restrict__ Y, int M)
{
    const int wave = threadIdx.x >> 5;       // 0..7 -> N tile
    const int lane = threadIdx.x & 31;
    const int half = lane >> 4;              // K-pair selector
    const int r    = lane & 15;              // row within tile
    const int m0   = blockIdx.x * 16;
    const int n0   = wave * 16;
    if (m0 >= M) return;                     // uniform per block

    const float* __restrict__ arow = A + (size_t)(m0 + r) * C;
    const float* __restrict__ brow = W + (size_t)(n0 + r) * C;   // B[k][n]=W[n][k]

    v8f c = {0.f, 0.f, 0.f, 0.f, 0.f, 0.f, 0.f, 0.f};
#pragma unroll
    for (int k0 = 0; k0 < C; k0 += 4) {
        v2f a = *(const v2f*)(arow + k0 + 2 * half);   // K = k0+2h, k0+2h+1
        v2f b = *(const v2f*)(brow + k0 + 2 * half);
        c = __builtin_amdgcn_wmma_f32_16x16x4_f32(
                false, a, false, b, (short)0, c, false, false);
    }
#pragma unroll
    for (int v = 0; v < 8; ++v)
        Y[(size_t)(m0 + v + 8 * half) * C + n0 + r] = c[v];
}

// ---------------------------------------------------------------------------
// Degrees: D[node] += hw[edge]; Bdeg[edge] += 1
// ---------------------------------------------------------------------------
__global__ void k_degrees(const int* __restrict__ nidx, const int* __restrict__ eidx,
                          const float* __restrict__ hw,
                          float* __restrict__ D, float* __restrict__ B, int nnz)
{
    int i = blockIdx.x * blockDim.x + threadIdx.x;
    if (i >= nnz) return;
    int e = eidx[i];
    atomicAdd(&D[nidx[i]], hw[e]);
    atomicAdd(&B[e], 1.0f);
}

__global__ void k_invert(float* __restrict__ v, int n)
{
    int i = blockIdx.x * blockDim.x + threadIdx.x;
    if (i >= n) return;
    float d = v[i];
    v[i] = (d > 0.f) ? (1.0f / d) : 0.0f;
}

// out[m*C + c] = bias[c]
__global__ void k_init_bias(float* __restrict__ out, const float* __restrict__ b, int total)
{
    int i = blockIdx.x * blockDim.x + threadIdx.x;
    if (i < total) out[i] = b[i & (C - 1)];
}

// ---------------------------------------------------------------------------
// node -> edge:  e[edge,:] += Binv[edge] * xw[node,:]
// block (32,8): 32 lanes x float4 = 128 channels, 8 incidences per block
// ---------------------------------------------------------------------------
__global__ __launch_bounds__(256) void k_scatter_n2e(
    const float* __restrict__ xw, const int* __restrict__ nidx,
    const int* __restrict__ eidx, const float* __restrict__ binv,
    float* __restrict__ e, int nnz)
{
    int inc = blockIdx.x * blockDim.y + threadIdx.y;
    if (inc >= nnz) return;
    int node = nidx[inc], edge = eidx[inc];
    float s = binv[edge];
    const float4 v = ((const float4*)(xw + (size_t)node * C))[threadIdx.x];
    float* dst = e + (size_t)edge * C + threadIdx.x * 4;
    atomicAdd(dst + 0, s * v.x);
    atomicAdd(dst + 1, s * v.y);
    atomicAdd(dst + 2, s * v.z);
    atomicAdd(dst + 3, s * v.w);
}

// edge -> node:  out[node,:] += Dinv[node] * e[edge,:]
__global__ __launch_bounds__(256) void k_scatter_e2n(
    const float* __restrict__ e, const int* __restrict__ nidx,
    const int* __restrict__ eidx, const float* __restrict__ dinv,
    float* __restrict__ out, int nnz)
{
    int inc = blockIdx.x * blockDim.y + threadIdx.y;
    if (inc >= nnz) return;
    int node = nidx[inc], edge = eidx[inc];
    float s = dinv[node];
    const float4 v = ((const float4*)(e + (size_t)edge * C))[threadIdx.x];
    float* dst = out + (size_t)node * C + threadIdx.x * 4;
    atomicAdd(dst + 0, s * v.x);
    atomicAdd(dst + 1, s * v.y);
    atomicAdd(dst + 2, s * v.z);
    atomicAdd(dst + 3, s * v.w);
}

// ---------------------------------------------------------------------------
// BatchNorm: one-pass sum / sumsq, per-channel. sums[0:128]=sum, [128:256]=sumsq
// ---------------------------------------------------------------------------
__global__ __launch_bounds__(128) void k_bn_stats(
    const float* __restrict__ h, float* __restrict__ sums, int M)
{
    int c  = threadIdx.x;
    int r0 = blockIdx.x * 128;
    int r1 = min(r0 + 128, M);
    float s = 0.f, s2 = 0.f;
    for (int r = r0; r < r1; ++r) {
        float v = h[(size_t)r * C + c];
        s += v; s2 += v * v;
    }
    atomicAdd(&sums[c], s);
    atomicAdd(&sums[C + c], s2);
}

__global__ void k_bn_apply(float* __restrict__ h, const float* __restrict__ sums,
                           const float* __restrict__ g, const float* __restrict__ bt,
                           int total, float invM)
{
    int i = blockIdx.x * blockDim.x + threadIdx.x;
    if (i >= total) return;
    int c = i & (C - 1);
    float mu  = sums[c] * invM;
    float var = sums[C + c] * invM - mu * mu;
    float inv = rsqrtf(var + BN_EPS);
    h[i] = g[c] * (h[i] - mu) * inv + bt[c];
}

// ---------------------------------------------------------------------------
// inputs: 0:x 1:hyperedge_index(int,[2,nnz]) 2:hyperedge_weight 3:W1 4:b1
//         5:gamma 6:beta 7:W2 8:b2       output: [N_NODES,128] f32
// ---------------------------------------------------------------------------
extern "C" void kernel_launch(void* const* d_in, const int* in_sizes, int n_in,
                              void* d_out, int out_size, void* d_ws, size_t ws_size,
                              hipStream_t stream)
{
    const float* x     = (const float*)d_in[0];
    const int*   hei   = (const int*)d_in[1];
    const float* hw    = (const float*)d_in[2];
    const float* W1    = (const float*)d_in[3];
    const float* b1    = (const float*)d_in[4];
    const float* gamma = (const float*)d_in[5];
    const float* beta  = (const float*)d_in[6];
    const float* W2    = (const float*)d_in[7];
    const float* b2    = (const float*)d_in[8];
    float*       out   = (float*)d_out;

    const int nnodes = in_sizes[0] / C;
    const int nnz    = in_sizes[1] / 2;
    const int nedges = in_sizes[2];
    const int* nidx  = hei;         // row 0
    const int* eidx  = hei + nnz;   // row 1

    // workspace carve-out
    char* p = (char*)d_ws;
    float* D    = (float*)p;  p += (size_t)nnodes * sizeof(float);
    float* Bd   = (float*)p;  p += (size_t)nedges * sizeof(float);
    float* xw   = (float*)p;  p += (size_t)nnodes * C * sizeof(float);
    float* e    = (float*)p;  p += (size_t)nedges * C * sizeof(float);
    float* h    = (float*)p;  p += (size_t)nnodes * C * sizeof(float);
    float* sums = (float*)p;  p += 2 * C * sizeof(float);

    const int totN = nnodes * C;

    // zero accumulators (every call: harness does not re-init ws)
    hipMemsetAsync(D,    0, (size_t)nnodes * sizeof(float), stream);
    hipMemsetAsync(Bd,   0, (size_t)nedges * sizeof(float), stream);
    hipMemsetAsync(e,    0, (size_t)nedges * C * sizeof(float), stream);
    hipMemsetAsync(sums, 0, 2 * C * sizeof(float), stream);

    // degrees -> inverse degrees
    k_degrees<<<(nnz + 255) / 256, 256, 0, stream>>>(nidx, eidx, hw, D, Bd, nnz);
    k_invert<<<(nnodes + 255) / 256, 256, 0, stream>>>(D, nnodes);
    k_invert<<<(nedges + 255) / 256, 256, 0, stream>>>(Bd, nedges);

    dim3 sblk(32, 8);
    int  sgrid = (nnz + 7) / 8;

    // ---- conv1: h = Dinv * scatter( Binv * scatter( x @ W1^T ) ) + b1 ----
    k_gemm_f32_wmma<<<nnodes / 16, 256, 0, stream>>>(x, W1, xw, nnodes);
    k_scatter_n2e<<<sgrid, sblk, 0, stream>>>(xw, nidx, eidx, Bd, e, nnz);
    k_init_bias<<<(totN + 255) / 256, 256, 0, stream>>>(h, b1, totN);
    k_scatter_e2n<<<sgrid, sblk, 0, stream>>>(e, nidx, eidx, D, h, nnz);

    // ---- batchnorm (training stats, biased variance) ----
    k_bn_stats<<<(nnodes + 127) / 128, 128, 0, stream>>>(h, sums, nnodes);
    k_bn_apply<<<(totN + 255) / 256, 256, 0, stream>>>(h, sums, gamma, beta,
                                                       totN, 1.0f / (float)nnodes);

    // ---- conv2 into d_out ----
    k_gemm_f32_wmma<<<nnodes / 16, 256, 0, stream>>>(h, W2, xw, nnodes);
    hipMemsetAsync(e, 0, (size_t)nedges * C * sizeof(float), stream);
    k_scatter_n2e<<<sgrid, sblk, 0, stream>>>(xw, nidx, eidx, Bd, e, nnz);
    k_init_bias<<<(totN + 255) / 256, 256, 0, stream>>>(out, b2, totN);
    k_scatter_e2n<<<sgrid, sblk, 0, stream>>>(e, nidx, eidx, D, out, nnz);
}